// GNN_22832046145825
// MI455X (gfx1250) — compile-verified
//
#include <hip/hip_runtime.h>

// f32 WMMA vector types (CDNA5 wave32): A/B of 16x16x4_f32 = 2 VGPRs/lane,
// C/D = 8 VGPRs/lane.
typedef __attribute__((ext_vector_type(2))) float v2f;
typedef __attribute__((ext_vector_type(8))) float v8f;

// ---------------------------------------------------------------------------
// Degree / normalization
// ---------------------------------------------------------------------------
__global__ void k_init_deg(float* __restrict__ deg, int n) {
    int i = blockIdx.x * blockDim.x + threadIdx.x;
    if (i < n) deg[i] = 1.0f;  // self-loop contributes 1 to every node
}

__global__ void k_accum_deg(const long long* __restrict__ dst,
                            float* __restrict__ deg, int e) {
    int i = blockIdx.x * blockDim.x + threadIdx.x;
    if (i < e) unsafeAtomicAdd(&deg[(int)dst[i]], 1.0f);
}

__global__ void k_finish_dinv(float* __restrict__ deg, int n) {
    int i = blockIdx.x * blockDim.x + threadIdx.x;
    if (i < n) {
        float d = deg[i];
        deg[i] = (d > 0.0f) ? rsqrtf(d) : 0.0f;
    }
}

// ---------------------------------------------------------------------------
// GEMM1: h1[N,16] = x[N,128] @ W1[128,16]  -- one wave per 16-row tile,
// K-loop of 32 x V_WMMA_F32_16X16X4_F32. N = 100000 = 6250*16 exactly,
// so EXEC is all-ones everywhere (WMMA requirement).
// ---------------------------------------------------------------------------
__global__ __launch_bounds__(32) void k_gemm1_wmma(const float* __restrict__ x,
                                                   const float* __restrict__ W1,
                                                   float* __restrict__ h1) {
    const int lane = threadIdx.x;       // 0..31
    const int m    = lane & 15;         // row within tile / B column
    const int khi  = lane >> 4;         // 0: K=0,1  1: K=2,3
    const int rowBase = blockIdx.x * 16;

    const float* xrow = x + (size_t)(rowBase + m) * 128 + 2 * khi;
    v8f c = {};
#pragma unroll
    for (int k0 = 0; k0 < 128; k0 += 4) {
        v2f a;
        a.x = xrow[k0 + 0];
        a.y = xrow[k0 + 1];
        v2f b;
        b.x = W1[(k0 + 2 * khi) * 16 + m];
        b.y = W1[(k0 + 2 * khi + 1) * 16 + m];
        c = __builtin_amdgcn_wmma_f32_16x16x4_f32(false, a, false, b,
                                                  (short)0, c, false, false);
    }
#pragma unroll
    for (int r = 0; r < 8; ++r) {
        int row = rowBase + r + 8 * khi;  // C/D layout: VGPR r -> M = r + 8*khi
        h1[row * 16 + m] = c[r];
    }
}

// ---------------------------------------------------------------------------
// GEMM2: h2[N,32] = z1[N,16] @ W2[16,32] -- two 16-col accumulators per wave.
// ---------------------------------------------------------------------------
__global__ __launch_bounds__(32) void k_gemm2_wmma(const float* __restrict__ z1,
                                                   const float* __restrict__ W2,
                                                   float* __restrict__ h2) {
    const int lane = threadIdx.x;
    const int m    = lane & 15;
    const int khi  = lane >> 4;
    const int rowBase = blockIdx.x * 16;

    const float* zr = z1 + (size_t)(rowBase + m) * 16 + 2 * khi;
    v8f c0 = {}, c1 = {};
#pragma unroll
    for (int k0 = 0; k0 < 16; k0 += 4) {
        v2f a;
        a.x = zr[k0 + 0];
        a.y = zr[k0 + 1];
        v2f b0, b1;
        b0.x = W2[(k0 + 2 * khi) * 32 + m];
        b0.y = W2[(k0 + 2 * khi + 1) * 32 + m];
        b1.x = W2[(k0 + 2 * khi) * 32 + 16 + m];
        b1.y = W2[(k0 + 2 * khi + 1) * 32 + 16 + m];
        c0 = __builtin_amdgcn_wmma_f32_16x16x4_f32(false, a, false, b0,
                                                   (short)0, c0, false, false);
        c1 = __builtin_amdgcn_wmma_f32_16x16x4_f32(false, a, false, b1,
                                                   (short)0, c1, false, false);
    }
#pragma unroll
    for (int r = 0; r < 8; ++r) {
        int row = rowBase + r + 8 * khi;
        h2[row * 32 + m]      = c0[r];
        h2[row * 32 + 16 + m] = c1[r];
    }
}

// ---------------------------------------------------------------------------
// Elementwise helpers
// ---------------------------------------------------------------------------
__global__ void k_zero(float* __restrict__ p, int n) {
    int i = blockIdx.x * blockDim.x + threadIdx.x;
    if (i < n) p[i] = 0.0f;
}

__global__ void k_relu_bias16(float* __restrict__ a, const float* __restrict__ b1,
                              int n) {  // n = N*16, in place: a = relu(a + b1[f])
    int i = blockIdx.x * blockDim.x + threadIdx.x;
    if (i < n) {
        float v = a[i] + b1[i & 15];
        a[i] = v > 0.0f ? v : 0.0f;
    }
}

__global__ void k_init_out_bias32(float* __restrict__ out,
                                  const float* __restrict__ b2, int n) {
    int i = blockIdx.x * blockDim.x + threadIdx.x;
    if (i < n) out[i] = b2[i & 31];  // n = N*32
}

// ---------------------------------------------------------------------------
// Scatter-add aggregation. One thread per (edge', 4-feature chunk).
// Edge indices e in [0,E) are real edges; e in [E, E+N) are self-loops.
// norm = dinv[src]*dinv[dst] computed on the fly (dinv table is L2-resident).
// ---------------------------------------------------------------------------
__global__ void k_scatter16(const long long* __restrict__ src,
                            const long long* __restrict__ dst,
                            const float* __restrict__ dinv,
                            const float* __restrict__ h,   // [N,16]
                            float* __restrict__ agg,       // [N,16]
                            int E, int Etot) {
    int t = blockIdx.x * blockDim.x + threadIdx.x;
    int e = t >> 2;
    if (e >= Etot) return;
    int c = (t & 3) * 4;
    int s, d;
    if (e < E) { s = (int)src[e]; d = (int)dst[e]; }
    else       { s = e - E; d = s; }
    float nrm = dinv[s] * dinv[d];
    const float4 hv = *(const float4*)(h + (size_t)s * 16 + c);
    float* o = agg + (size_t)d * 16 + c;
    unsafeAtomicAdd(o + 0, hv.x * nrm);
    unsafeAtomicAdd(o + 1, hv.y * nrm);
    unsafeAtomicAdd(o + 2, hv.z * nrm);
    unsafeAtomicAdd(o + 3, hv.w * nrm);
}

__global__ void k_scatter32(const long long* __restrict__ src,
                            const long long* __restrict__ dst,
                            const float* __restrict__ dinv,
                            const float* __restrict__ h,   // [N,32]
                            float* __restrict__ agg,       // [N,32] = d_out
                            int E, int Etot) {
    int t = blockIdx.x * blockDim.x + threadIdx.x;
    int e = t >> 3;
    if (e >= Etot) return;
    int c = (t & 7) * 4;
    int s, d;
    if (e < E) { s = (int)src[e]; d = (int)dst[e]; }
    else       { s = e - E; d = s; }
    float nrm = dinv[s] * dinv[d];
    const float4 hv = *(const float4*)(h + (size_t)s * 32 + c);
    float* o = agg + (size_t)d * 32 + c;
    unsafeAtomicAdd(o + 0, hv.x * nrm);
    unsafeAtomicAdd(o + 1, hv.y * nrm);
    unsafeAtomicAdd(o + 2, hv.z * nrm);
    unsafeAtomicAdd(o + 3, hv.w * nrm);
}

// ---------------------------------------------------------------------------
// Launch
// ---------------------------------------------------------------------------
extern "C" void kernel_launch(void* const* d_in, const int* in_sizes, int n_in,
                              void* d_out, int out_size, void* d_ws, size_t ws_size,
                              hipStream_t stream) {
    const float*     x   = (const float*)d_in[0];
    const long long* ei  = (const long long*)d_in[1];   // int64 edge_index [2,E]
    const float*     W1  = (const float*)d_in[2];
    const float*     b1  = (const float*)d_in[3];
    const float*     W2  = (const float*)d_in[4];
    const float*     b2  = (const float*)d_in[5];
    float*           out = (float*)d_out;

    const int N = in_sizes[0] / 128;   // 100000
    const int E = in_sizes[1] / 2;     // 3200000
    const int Etot = E + N;            // edges + self-loops

    const long long* src = ei;
    const long long* dst = ei + E;

    // Workspace layout (floats): dinv[N] | h1[N*16] | agg1[N*16] | h2[N*32]
    float* dinv = (float*)d_ws;
    float* h1   = dinv + N;
    float* agg1 = h1 + (size_t)N * 16;
    float* h2   = agg1 + (size_t)N * 16;

    const int B = 256;
    auto blocks = [&](long long n) { return (unsigned)((n + B - 1) / B); };

    // 1) symmetric normalization: deg -> dinv (in place)
    k_init_deg<<<blocks(N), B, 0, stream>>>(dinv, N);
    k_accum_deg<<<blocks(E), B, 0, stream>>>(dst, dinv, E);
    k_finish_dinv<<<blocks(N), B, 0, stream>>>(dinv, N);

    // 2) h1 = x @ W1   (WMMA f32, one wave per 16-row tile; N % 16 == 0)
    k_gemm1_wmma<<<N / 16, 32, 0, stream>>>(x, W1, h1);

    // 3) agg1 = scatter-add(norm * h1[src]) ; then relu(agg1 + b1) in place
    k_zero<<<blocks((long long)N * 16), B, 0, stream>>>(agg1, N * 16);
    k_scatter16<<<blocks((long long)Etot * 4), B, 0, stream>>>(src, dst, dinv,
                                                               h1, agg1, E, Etot);
    k_relu_bias16<<<blocks((long long)N * 16), B, 0, stream>>>(agg1, b1, N * 16);

    // 4) h2 = z1 @ W2  (WMMA f32)
    k_gemm2_wmma<<<N / 16, 32, 0, stream>>>(agg1, W2, h2);

    // 5) out = b2 + scatter-add(norm * h2[src])  (accumulate directly in d_out)
    k_init_out_bias32<<<blocks((long long)N * 32), B, 0, stream>>>(out, b2, N * 32);
    k_scatter32<<<blocks((long long)Etot * 8), B, 0, stream>>>(src, dst, dinv,
                                                               h2, out, E, Etot);
}